// TensorTrainGaussian2D_58282706206886
// MI455X (gfx1250) — compile-verified
//
#include <hip/hip_runtime.h>

// TensorTrain 2D Gaussian mixture likelihood for gfx1250 (MI455X).
// Exponent arguments computed via V_WMMA_F32_16X16X4_F32 (rank-3 feature GEMM),
// pdf via hardware v_exp_f32 (base-2, log2e folded into precomputed G tables),
// reductions via per-lane-weight FMAs + ds_bpermute broadcasts.
// Outer loops kept rolled + waves_per_eu cap to avoid VGPR blowup/spills.

typedef float v2f __attribute__((ext_vector_type(2)));
typedef float v8f __attribute__((ext_vector_type(8)));

#define KDIM 32
#define JTOT (KDIM * KDIM)   // 1024 (k-pair plane)
#define PAD  16              // LDS bank stagger

__device__ __forceinline__ float fast_exp2(float x) {
#if __has_builtin(__builtin_amdgcn_exp2f)
  return __builtin_amdgcn_exp2f(x);      // v_exp_f32
#else
  return exp2f(x);
#endif
}
__device__ __forceinline__ float fast_log2(float x) {
#if __has_builtin(__builtin_amdgcn_logf)
  return __builtin_amdgcn_logf(x);       // v_log_f32
#else
  return log2f(x);
#endif
}

__global__ __launch_bounds__(256)
__attribute__((amdgpu_waves_per_eu(4)))
void tt_gauss2d_wmma(
    const float* __restrict__ X,     // [N,2]
    const float* __restrict__ Wk0,   // [32]
    const float* __restrict__ W1,    // Wk1k0 [32,32]
    const float* __restrict__ W2,    // Wk2k1 [32,32]
    const float* __restrict__ mu,    // [32,32]
    const float* __restrict__ sigma, // [32,32]
    float* __restrict__ out,         // [N]
    int n_groups)                    // N/16
{
  __shared__ float sG0[JTOT + PAD];
  __shared__ float sG1[JTOT + PAD];
  __shared__ float sG2[JTOT + PAD];
  __shared__ float sW1[JTOT + PAD];
  __shared__ float sW2[JTOT + PAD];
  __shared__ float sWk0[KDIM];

  const int tid = (int)threadIdx.x;
  const float LOG2E = 1.4426950408889634f;

  // --- stage parameters / precompute exponent-polynomial tables (base-2) ---
  for (int j = tid; j < JTOT; j += 256) {
    float m_ = mu[j], s_ = sigma[j];
    float inv_s  = 1.0f / s_;
    float inv_s2 = inv_s * inv_s;
    sG0[j] = -0.5f * LOG2E * inv_s2;                 // coef of x^2
    sG1[j] = LOG2E * m_ * inv_s2;                    // coef of x
    sG2[j] = fast_log2(0.3989422804014327f * inv_s)  // log2(1/(sigma*sqrt(2pi)))
             - 0.5f * LOG2E * m_ * m_ * inv_s2;      // constant term
    sW1[j] = W1[j];
    sW2[j] = W2[j];
  }
  if (tid < KDIM) sWk0[tid] = Wk0[tid];
  __syncthreads();

  const int  lane = tid & 31;
  const int  wave = tid >> 5;
  const int  col  = lane & 15;          // WMMA D column / sample-within-16
  const bool lo   = lane < 16;
  const int  hsel = lane >> 4;          // which 8-sample half this lane holds
  // B operand (4x16 f32): VGPR0 lanes0-15 = K0 row (G0), lanes16-31 = K2 row (G2)
  const float* BxBase = lo ? sG0 : sG2;

#pragma unroll 1
  for (int g = (int)blockIdx.x * 8 + wave; g < n_groups; g += (int)gridDim.x * 8) {
    // A operand (16x4 f32): lanes0-15 {K0,K1} = {x^2, x}; lanes16-31 {K2,K3} = {1, 0}
    float x0 = 0.0f, x1 = 0.0f;
    if (lo) {
      int n0 = g * 16 + col;
      x0 = X[2 * n0 + 0];
      x1 = X[2 * n0 + 1];
    }
    v2f A0, A1;
    A1.x = lo ? x1 * x1 : 1.0f;  A1.y = x1;   // x==0 on upper lanes -> K3 = 0
    A0.x = lo ? x0 * x0 : 1.0f;  A0.y = x0;

    // ---------- Pass 1: inner[m,k1] = sum_k2 W2[k2,k1] * P1[m,k2,k1] ----------
    float innerAcc[2][8];
#pragma unroll
    for (int h = 0; h < 2; ++h)
#pragma unroll
      for (int v = 0; v < 8; ++v) innerAcc[h][v] = 0.0f;

#pragma unroll 1
    for (int a = 0; a < KDIM; ++a) {            // a = k2  (keep rolled!)
#pragma unroll
      for (int h = 0; h < 2; ++h) {             // k1 half: k1 = h*16 + col
        int j = a * KDIM + h * 16 + col;        // flat (k2,k1)
        v2f B;
        B.x = BxBase[j];
        float g1v = sG1[j];
        B.y = lo ? g1v : 0.0f;
        v8f c = {};
        v8f d = __builtin_amdgcn_wmma_f32_16x16x4_f32(
            false, A1, false, B, (short)0, c, false, false);
        float w = sW2[j];                       // W2[k2, k1], same flattening
        float p[8];
#pragma unroll
        for (int v = 0; v < 8; ++v) p[v] = fast_exp2(d[v]);
#pragma unroll
        for (int v = 0; v < 8; ++v)
          innerAcc[h][v] = fmaf(p[v], w, innerAcc[h][v]);
      }
    }

    // ---- Pass 2: mid[m,k0] = sum_k1 W1[k1,k0] * P0[m,k1,k0] * inner[m,k1] ----
    float midAcc[2][8];
#pragma unroll
    for (int h = 0; h < 2; ++h)
#pragma unroll
      for (int v = 0; v < 8; ++v) midAcc[h][v] = 0.0f;

#pragma unroll
    for (int half = 0; half < 2; ++half) {      // which innerAcc bank holds k1
#pragma unroll 1
      for (int a0 = 0; a0 < 16; ++a0) {         // keep rolled!
        int a = half * 16 + a0;                 // a = k1
        int srcLane = (lane & 16) | a0;         // broadcast within each 16-lane half
        float ib[8];
#pragma unroll
        for (int v = 0; v < 8; ++v)
          ib[v] = __shfl(innerAcc[half][v], srcLane, 32);  // ds_bpermute
#pragma unroll
        for (int h = 0; h < 2; ++h) {           // k0 half: k0 = h*16 + col
          int j = a * KDIM + h * 16 + col;      // flat (k1,k0)
          v2f B;
          B.x = BxBase[j];
          float g1v = sG1[j];
          B.y = lo ? g1v : 0.0f;
          v8f c = {};
          v8f d = __builtin_amdgcn_wmma_f32_16x16x4_f32(
              false, A0, false, B, (short)0, c, false, false);
          float w = sW1[j];                     // W1[k1, k0]
          float p[8];
#pragma unroll
          for (int v = 0; v < 8; ++v) p[v] = fast_exp2(d[v]) * w;
#pragma unroll
          for (int v = 0; v < 8; ++v)
            midAcc[h][v] = fmaf(p[v], ib[v], midAcc[h][v]);
        }
      }
    }

    // ---------- Final: log(sum_k0 Wk0[k0] * mid[m,k0]) ----------
    float wk0a = sWk0[col];
    float wk0b = sWk0[16 + col];
#pragma unroll
    for (int v = 0; v < 8; ++v) {
      float s = midAcc[0][v] * wk0a + midAcc[1][v] * wk0b;
      s += __shfl_xor(s, 1, 32);   // butterfly within each 16-lane half
      s += __shfl_xor(s, 2, 32);
      s += __shfl_xor(s, 4, 32);
      s += __shfl_xor(s, 8, 32);
      if (col == 0)
        out[g * 16 + hsel * 8 + v] = fast_log2(s) * 0.6931471805599453f;
    }
  }
}

extern "C" void kernel_launch(void* const* d_in, const int* in_sizes, int n_in,
                              void* d_out, int out_size, void* d_ws, size_t ws_size,
                              hipStream_t stream) {
  const float* X     = (const float*)d_in[0];
  const float* Wk0   = (const float*)d_in[1];
  const float* W1    = (const float*)d_in[2];  // Wk1k0
  const float* W2    = (const float*)d_in[3];  // Wk2k1
  const float* mu    = (const float*)d_in[4];
  const float* sigma = (const float*)d_in[5];
  float* out = (float*)d_out;

  int N = in_sizes[0] / 2;
  int n_groups = (N + 15) / 16;               // 16 samples per wave per group
  int blocks = (n_groups + 7) / 8;            // 8 waves (256 threads) per block
  if (blocks < 1) blocks = 1;
  if (blocks > 4096) blocks = 4096;           // grid-stride handles the rest

  tt_gauss2d_wmma<<<blocks, 256, 0, stream>>>(X, Wk0, W1, W2, mu, sigma, out,
                                              n_groups);
}